// CpGenerateExport_43645457662696
// MI455X (gfx1250) — compile-verified
//
#include <hip/hip_runtime.h>
#include <hip/hip_bf16.h>
#include <math.h>

// ---------------------------------------------------------------------------
// CDNA5 (gfx1250) implementation of the talker code-predict generate loop.
//
// MI455X reasoning:
//  * Every matmul is a GEMV (M=1). Weight bytes dominate (96MB/step f32,
//    ~222MB working set vs 192MB L2) -> bandwidth bound; matrix MACs free.
//    Use V_WMMA_F32_16X16X4_F32 to consume f32 weights directly.
//  * One wave32 owns a 16-wide output tile; K swept in batches of 16
//    (4 WMMAs), with all loads for a batch issued as a clause before the
//    WMMA burst so LOADcnt latency overlaps the matrix pipe.
//  * K/N are template constants -> strides fold into global_load immediate
//    offsets; no 64-bit multiplies in the inner loop.
//  * A-matrix row-0 masking done with a VALU multiply (no EXEC divergence).
//  * Sequential sampling dependency -> host-side launch chain (graph-safe:
//    kernel launches + hipMemcpyAsync D2D only).
// ---------------------------------------------------------------------------

typedef float v2f __attribute__((ext_vector_type(2)));
typedef float v8f __attribute__((ext_vector_type(8)));

#define D_MODEL 1024
#define FF_DIM  4096
#define N_HEADS 16
#define HEAD_D  64
#define VOCAB   2048
#define N_GRP   15
#define N_TOPK  50
#define MAX_T   16

// ---------------------------------------------------------------------------
// y[0:N) = act( x[0:K) @ W[K x N]  (+ addv) )   -- one wave per 16 columns.
// ---------------------------------------------------------------------------
template <int K, int N>
__global__ void __launch_bounds__(128)
gemv_xw_wmma(const float* __restrict__ x, const float* __restrict__ W,
             const float* __restrict__ addv, float* __restrict__ y,
             int act_gelu) {
  __shared__ float xs[K];
  const int tid = threadIdx.x;
  for (int i = tid; i < K; i += 128) xs[i] = x[i];
  __syncthreads();

  const int wave = tid >> 5;
  const int lane = tid & 31;
  const int nb   = (blockIdx.x * 4 + wave) * 16;       // 16-column tile
  const int col  = nb + (lane & 15);                   // B: lane -> column
  const int ko   = (lane >> 4) << 1;                   // K sub-pair (0 or 2)
  const float rowm = ((lane & 15) == 0) ? 1.0f : 0.0f; // A row-0 lane mask

  // Wp points at W[ko][col]; marches 16 K-rows per outer iteration.
  const float* __restrict__ Wp = W + (size_t)ko * N + col;
  v8f acc = {};

  for (int kbase = 0; kbase < K; kbase += 16) {
    __builtin_prefetch(Wp + 64 * N, 0, 1);   // 4 batches ahead

    v2f a[4], b[4];
#pragma unroll
    for (int j = 0; j < 4; j++) {
      // Unpredicated LDS pair load (8B aligned), masked by VALU multiply.
      const v2f xv = *reinterpret_cast<const v2f*>(&xs[kbase + 4 * j + ko]);
      a[j].x = xv.x * rowm;
      a[j].y = xv.y * rowm;
      // Constant strides: (4j)*N and (4j+1)*N fold into load offsets.
      b[j].x = Wp[(4 * j) * N];
      b[j].y = Wp[(4 * j + 1) * N];
    }
#pragma unroll
    for (int j = 0; j < 4; j++) {
      acc = __builtin_amdgcn_wmma_f32_16x16x4_f32(
          /*neg_a=*/false, a[j], /*neg_b=*/false, b[j],
          /*c_mod=*/(short)0, acc, /*reuse_a=*/false, /*reuse_b=*/false);
    }
    Wp += 16 * (size_t)N;
  }

  if (lane < 16) {                 // D row 0 lives in VGPR0 of lanes 0..15
    float r = acc[0];
    if (addv) r += addv[col];
    if (act_gelu) {                // jax.nn.gelu tanh approximation
      const float c = r + 0.044715f * r * r * r;
      r = 0.5f * r * (1.0f + tanhf(0.7978845608028654f * c));
    }
    y[col] = r;
  }
}

// ---------------------------------------------------------------------------
// y[v] = H[v,:] . x  (vocab-head GEMV) -- 16 rows per wave via WMMA.
// A = 16 rows of H (16x4), B column 0 = x chunk, D column 0 = results.
// ---------------------------------------------------------------------------
template <int K>
__global__ void __launch_bounds__(128)
gemv_rows_wmma(const float* __restrict__ x, const float* __restrict__ H,
               float* __restrict__ y) {
  __shared__ float xs[K];
  const int tid = threadIdx.x;
  for (int i = tid; i < K; i += 128) xs[i] = x[i];
  __syncthreads();

  const int wave = tid >> 5;
  const int lane = tid & 31;
  const int vb   = (blockIdx.x * 4 + wave) * 16;       // 16-row tile
  const int m    = lane & 15;                          // A: lane -> row
  const int ko   = (lane >> 4) << 1;                   // K sub-pair
  const float colm = ((lane & 15) == 0) ? 1.0f : 0.0f; // B col-0 lane mask

  const float* __restrict__ Hp = H + (size_t)(vb + m) * K + ko;
  v8f acc = {};

  for (int kbase = 0; kbase < K; kbase += 16) {
    __builtin_prefetch(Hp + 64, 0, 1);

    v2f a[4], b[4];
#pragma unroll
    for (int j = 0; j < 4; j++) {
      a[j] = *reinterpret_cast<const v2f*>(Hp + 4 * j);   // contiguous pair
      const v2f xv = *reinterpret_cast<const v2f*>(&xs[kbase + 4 * j + ko]);
      b[j].x = xv.x * colm;
      b[j].y = xv.y * colm;
    }
#pragma unroll
    for (int j = 0; j < 4; j++) {
      acc = __builtin_amdgcn_wmma_f32_16x16x4_f32(
          false, a[j], false, b[j], (short)0, acc, false, false);
    }
    Hp += 16;
  }

  // D column 0: lane 0 holds M=0..7 (VGPR0..7), lane 16 holds M=8..15.
  if (lane == 0) {
#pragma unroll
    for (int i = 0; i < 8; i++) y[vb + i] = acc[i];
  }
  if (lane == 16) {
#pragma unroll
    for (int i = 0; i < 8; i++) y[vb + 8 + i] = acc[i];
  }
}

// ---------------------------------------------------------------------------
// RMSNorm: y = x * rsqrt(mean(x^2)+1e-6) * w
// ---------------------------------------------------------------------------
__global__ void __launch_bounds__(256)
rms_kernel(const float* __restrict__ x, const float* __restrict__ w,
           float* __restrict__ y, int D) {
  __shared__ float red[256];
  const int t = threadIdx.x;
  float s = 0.0f;
  for (int i = t; i < D; i += 256) { const float v = x[i]; s += v * v; }
  red[t] = s;
  __syncthreads();
  for (int o = 128; o > 0; o >>= 1) {
    if (t < o) red[t] += red[t + o];
    __syncthreads();
  }
  const float inv = rsqrtf(red[0] / (float)D + 1e-6f);
  for (int i = t; i < D; i += 256) y[i] = x[i] * inv * w[i];
}

// ---------------------------------------------------------------------------
// Per-head RoPE + causal attention over the (<=16 entry) KV cache.
// grid = NH heads, block = HEAD_D threads.
// ---------------------------------------------------------------------------
__global__ void __launch_bounds__(64)
attn_kernel(const float* __restrict__ q, const float* __restrict__ k,
            const float* __restrict__ v, float* __restrict__ kcache,
            float* __restrict__ vcache, float* __restrict__ out,
            int layer, int pos) {
  const int h = blockIdx.x;
  const int t = threadIdx.x;                    // 0..63
  const int base = h * HEAD_D + t;

  const float qv = q[base], kv = k[base], vv = v[base];

  // RoPE (half = 32): pair (i, i+32)
  const int i = t & 31;
  const float inv = powf(10000.0f, -(float)i / 32.0f);
  const float ang = (float)pos * inv;
  const float c = cosf(ang), sn = sinf(ang);
  float qr, kr;
  if (t < 32) { qr = qv * c - q[base + 32] * sn; kr = kv * c - k[base + 32] * sn; }
  else        { qr = qv * c + q[base - 32] * sn; kr = kv * c + k[base - 32] * sn; }

  const size_t slot = ((size_t)(layer * MAX_T + pos) * N_HEADS + h) * HEAD_D + t;
  kcache[slot] = kr;
  vcache[slot] = vv;

  __shared__ float red[64];
  __shared__ float sc[MAX_T + 1];

  for (int s = 0; s <= pos; s++) {
    const float kk = (s == pos)
        ? kr
        : kcache[((size_t)(layer * MAX_T + s) * N_HEADS + h) * HEAD_D + t];
    red[t] = qr * kk;
    __syncthreads();
    for (int o = 32; o > 0; o >>= 1) {
      if (t < o) red[t] += red[t + o];
      __syncthreads();
    }
    if (t == 0) sc[s] = red[0] * 0.125f;        // 1/sqrt(64)
    __syncthreads();
  }

  if (t == 0) {
    float mx = -1e30f;
    for (int s = 0; s <= pos; s++) mx = fmaxf(mx, sc[s]);
    float sum = 0.0f;
    for (int s = 0; s <= pos; s++) { sc[s] = expf(sc[s] - mx); sum += sc[s]; }
    const float isum = 1.0f / sum;
    for (int s = 0; s <= pos; s++) sc[s] *= isum;
  }
  __syncthreads();

  float acc = 0.0f;
  for (int s = 0; s <= pos; s++) {
    const float vs = (s == pos)
        ? vv
        : vcache[((size_t)(layer * MAX_T + s) * N_HEADS + h) * HEAD_D + t];
    acc += sc[s] * vs;
  }
  out[base] = acc;
}

// ---------------------------------------------------------------------------
// Top-k(50) sample: argmax-sweep selection, softmax/temperature, CDF pick,
// then gather embedding row, accumulate embed_sum, seed next-step hidden.
// ---------------------------------------------------------------------------
__global__ void __launch_bounds__(256)
sample_kernel(const float* __restrict__ logits, const float* __restrict__ temp,
              const float* __restrict__ uniforms, int g,
              const float* __restrict__ embeds, float* __restrict__ embed_sum,
              float* __restrict__ hbuf, float* __restrict__ out_codes) {
  __shared__ float sv[VOCAB];
  __shared__ float rv[256];
  __shared__ int   ri[256];
  __shared__ float topv[N_TOPK];
  __shared__ int   topi[N_TOPK];
  __shared__ int   code_sh;

  const int t = threadIdx.x;
  for (int i2 = t; i2 < VOCAB; i2 += 256) sv[i2] = logits[i2];
  __syncthreads();

  for (int r = 0; r < N_TOPK; r++) {
    float bv = -3.0e38f;
    int   bi = 0x7fffffff;
    for (int i2 = t; i2 < VOCAB; i2 += 256) {
      const float vv = sv[i2];
      if (vv > bv || (vv == bv && i2 < bi)) { bv = vv; bi = i2; }
    }
    rv[t] = bv; ri[t] = bi;
    __syncthreads();
    for (int o = 128; o > 0; o >>= 1) {
      if (t < o) {
        if (rv[t + o] > rv[t] || (rv[t + o] == rv[t] && ri[t + o] < ri[t])) {
          rv[t] = rv[t + o];
          ri[t] = ri[t + o];
        }
      }
      __syncthreads();
    }
    if (t == 0) { topv[r] = rv[0]; topi[r] = ri[0]; sv[ri[0]] = -3.0e38f; }
    __syncthreads();
  }

  if (t == 0) {
    const float st = fmaxf(temp[0], 1e-5f);
    const float mx = topv[0];                  // descending -> top1 is max
    float p[N_TOPK];
    float sum = 0.0f;
    for (int r = 0; r < N_TOPK; r++) { p[r] = expf((topv[r] - mx) / st); sum += p[r]; }
    float u = uniforms[g];
    u = fminf(fmaxf(u, 1e-6f), 1.0f - 1e-6f);
    float cdf = 0.0f;
    int choice = 0;
    bool found = false;
    for (int r = 0; r < N_TOPK; r++) {
      cdf += p[r] / sum;
      if (!found && cdf >= u) { choice = r; found = true; }
    }
    const int code = topi[choice];
    code_sh = code;
    out_codes[g] = (float)code;
  }
  __syncthreads();

  const int code = code_sh;
  const float* erow = embeds + ((size_t)g * VOCAB + code) * D_MODEL;
  for (int i2 = t; i2 < D_MODEL; i2 += 256) {
    const float e = erow[i2];
    embed_sum[i2] += e;
    hbuf[i2] = e;
  }
}

// ---------------------------------------------------------------------------
// Host orchestration (graph-capture safe: launches + D2D async copies only).
// ---------------------------------------------------------------------------
extern "C" void kernel_launch(void* const* d_in, const int* in_sizes, int n_in,
                              void* d_out, int out_size, void* d_ws, size_t ws_size,
                              hipStream_t stream) {
  (void)in_sizes; (void)n_in; (void)out_size; (void)ws_size;

  const float* talker  = (const float*)d_in[0];   // (1,1,D)
  const float* code0   = (const float*)d_in[1];   // (1,1,D)
  const float* temp    = (const float*)d_in[2];   // (1,)
  const float* unif    = (const float*)d_in[3];   // (G,)
  const float* heads   = (const float*)d_in[4];   // (G,V,D)
  const float* embeds  = (const float*)d_in[5];   // (G,V,D)
  const float* ln1     = (const float*)d_in[6];   // (NL,D)
  const float* ln2     = (const float*)d_in[7];   // (NL,D)
  const float* lnf     = (const float*)d_in[8];   // (D,)
  const float* wq      = (const float*)d_in[9];   // (NL,D,D)
  const float* wk      = (const float*)d_in[10];
  const float* wv      = (const float*)d_in[11];
  const float* wo      = (const float*)d_in[12];
  const float* w1      = (const float*)d_in[13];  // (NL,D,FF)
  const float* w2      = (const float*)d_in[14];  // (NL,FF,D)
  float* out = (float*)d_out;                     // [0..14]=codes, [15..1038]=embed_sum

  // Workspace layout (floats): ~316KB total.
  float* ws  = (float*)d_ws;
  float* h0  = ws;                 // 1024  hidden state (ping)
  float* h1  = ws + 1024;          // 1024  hidden state (pong)
  float* xb  = ws + 2048;          // 1024  rms output
  float* qb  = ws + 3072;          // 1024
  float* kb  = ws + 4096;          // 1024
  float* vb  = ws + 5120;          // 1024
  float* ab  = ws + 6144;          // 1024  attention out
  float* ffb = ws + 7168;          // 4096  MLP intermediate
  float* hid = ws + 11264;         // 1024  final-rms hidden
  float* lg  = ws + 12288;         // 2048  logits
  float* es  = ws + 14336;         // 1024  embed_sum
  float* kc  = ws + 15360;         // 2*16*16*64 = 32768 K cache
  float* vc  = ws + 48128;         // 32768 V cache

  auto step = [&](int pos) {
    float* hcur = h0;
    float* hnxt = h1;
    for (int l = 0; l < 2; l++) {
      const size_t dd = (size_t)l * D_MODEL * D_MODEL;
      const size_t df = (size_t)l * D_MODEL * FF_DIM;
      rms_kernel<<<1, 256, 0, stream>>>(hcur, ln1 + l * D_MODEL, xb, D_MODEL);
      gemv_xw_wmma<D_MODEL, D_MODEL><<<16, 128, 0, stream>>>(xb, wq + dd, nullptr, qb, 0);
      gemv_xw_wmma<D_MODEL, D_MODEL><<<16, 128, 0, stream>>>(xb, wk + dd, nullptr, kb, 0);
      gemv_xw_wmma<D_MODEL, D_MODEL><<<16, 128, 0, stream>>>(xb, wv + dd, nullptr, vb, 0);
      attn_kernel<<<N_HEADS, HEAD_D, 0, stream>>>(qb, kb, vb, kc, vc, ab, l, pos);
      gemv_xw_wmma<D_MODEL, D_MODEL><<<16, 128, 0, stream>>>(ab, wo + dd, hcur, hnxt, 0);
      { float* tmpp = hcur; hcur = hnxt; hnxt = tmpp; }
      rms_kernel<<<1, 256, 0, stream>>>(hcur, ln2 + l * D_MODEL, xb, D_MODEL);
      gemv_xw_wmma<D_MODEL, FF_DIM><<<64, 128, 0, stream>>>(xb, w1 + df, nullptr, ffb, 1);
      gemv_xw_wmma<FF_DIM, D_MODEL><<<16, 128, 0, stream>>>(ffb, w2 + df, hcur, hnxt, 0);
      { float* tmpp = hcur; hcur = hnxt; hnxt = tmpp; }
    }
    // 4 swaps per step -> hcur == h0 here.
    rms_kernel<<<1, 256, 0, stream>>>(hcur, lnf, hid, D_MODEL);
  };

  // Prefill: causal => process the two tokens sequentially (identical result).
  hipMemcpyAsync(h0, talker, D_MODEL * sizeof(float), hipMemcpyDeviceToDevice, stream);
  step(0);
  hipMemcpyAsync(h0, code0, D_MODEL * sizeof(float), hipMemcpyDeviceToDevice, stream);
  step(1);
  hipMemcpyAsync(es, code0, D_MODEL * sizeof(float), hipMemcpyDeviceToDevice, stream);

  // Generate loop.
  for (int g = 0; g < N_GRP; g++) {
    gemv_rows_wmma<D_MODEL><<<VOCAB / 64, 128, 0, stream>>>(
        hid, heads + (size_t)g * VOCAB * D_MODEL, lg);
    sample_kernel<<<1, 256, 0, stream>>>(lg, temp, unif, g, embeds, es, h0, out);
    if (g + 1 < N_GRP) step(g + 2);
  }

  hipMemcpyAsync(out + N_GRP, es, D_MODEL * sizeof(float),
                 hipMemcpyDeviceToDevice, stream);
}